// Detect_59794534695140
// MI455X (gfx1250) — compile-verified
//
#include <hip/hip_runtime.h>
#include <hip/hip_bf16.h>

typedef __attribute__((ext_vector_type(2))) float v2f;
typedef __attribute__((ext_vector_type(8))) float v8f;

#define TOPK   200
#define PADK   208            // 13 * 16, WMMA tile padding
#define NTILE  13
#define NT     256            // 8 wave32 per block
#define CONF_T 0.05f
#define NMS_T  0.45f
#define VAR0   0.1f
#define VAR1   0.2f

__global__ __launch_bounds__(NT)
void detect_nms_kernel(const float* __restrict__ loc,
                       const float* __restrict__ conf,
                       const float* __restrict__ anchors,
                       float* __restrict__ out,
                       int B, int A, int C) {
    const int t  = threadIdx.x;
    const int bc = blockIdx.x;            // b * C + c
    const int b  = bc / C;
    const int c  = bc % C;

    __shared__ unsigned hist[256];
    __shared__ unsigned top_key[NT];
    __shared__ int      top_idx[NT];
    __shared__ __align__(16) float lds_loc[PADK * 4];   // async-staged loc rows
    __shared__ __align__(16) float lds_anc[PADK * 4];   // async-staged anchor rows
    __shared__ __align__(16) float4 bbx[PADK];          // decoded boxes (x1,y1,x2,y2) AoS
    __shared__ float barea[PADK], bscore[PADK];
    __shared__ unsigned sup[PADK][7];     // suppression bitmask rows (208 bits -> 7 words)
    __shared__ unsigned keepw[7];
    __shared__ unsigned s_prefix, s_k, cnt_gt, cnt_eq;

    const float* confb = conf + (size_t)b * (size_t)A * (size_t)C;

    // ---------------- radix select: exact 200th-largest key ----------------
    // key = order-preserving uint of score if score > CONF_T else 0
    if (t == 0) { s_prefix = 0u; s_k = TOPK; }
    __syncthreads();

    for (int pass = 0; pass < 4; ++pass) {
        hist[t] = 0u;
        __syncthreads();
        const unsigned prefix = s_prefix;
        const int shift = 24 - 8 * pass;
        const int hs = shift + 8;
        for (int a = t; a < A; a += NT) {
            if (a + NT < A)
                __builtin_prefetch(&confb[(size_t)(a + NT) * C + c], 0, 1);
            float s = confb[(size_t)a * C + c];
            unsigned key = (s > CONF_T) ? __float_as_uint(s) : 0u;
            if (hs >= 32 || (key >> hs) == prefix)
                atomicAdd(&hist[(key >> shift) & 255u], 1u);
        }
        __syncthreads();
        if (t == 0) {
            unsigned k = s_k, cum = 0u; int sel = 0;
            for (int bin = 255; bin >= 0; --bin) {
                unsigned h = hist[bin];
                if (cum + h >= k) { sel = bin; break; }
                cum += h;
            }
            s_k = k - cum;
            s_prefix = (prefix << 8) | (unsigned)sel;
        }
        __syncthreads();
    }

    const unsigned T   = s_prefix;        // exact threshold key
    const unsigned kk  = s_k;             // how many ties (== T) to take
    const unsigned ngt = TOPK - kk;       // count of keys strictly greater
    if (t == 0) { cnt_gt = 0u; cnt_eq = 0u; }
    __syncthreads();

    // ---------------- compaction of top-200 candidates ----------------
    for (int a = t; a < A; a += NT) {
        float s = confb[(size_t)a * C + c];
        unsigned key = (s > CONF_T) ? __float_as_uint(s) : 0u;
        if (key > T) {
            unsigned p = atomicAdd(&cnt_gt, 1u);
            if (p < TOPK) { top_key[p] = key; top_idx[p] = a; }
        } else if (key == T) {
            unsigned p = atomicAdd(&cnt_eq, 1u);
            if (p < kk) { top_key[ngt + p] = key; top_idx[ngt + p] = a; }
        }
    }
    __syncthreads();
    if (t >= TOPK) { top_key[t] = 0u; top_idx[t] = 0; }
    __syncthreads();

    // ---------------- bitonic sort (descending), 256 elems ----------------
    for (unsigned k = 2; k <= NT; k <<= 1) {
        for (unsigned j = k >> 1; j > 0; j >>= 1) {
            unsigned ixj = t ^ j;
            if (ixj > (unsigned)t) {
                bool desc = ((t & k) == 0);
                unsigned k0 = top_key[t], k1 = top_key[ixj];
                if (desc ? (k0 < k1) : (k0 > k1)) {
                    top_key[t] = k1; top_key[ixj] = k0;
                    int i0 = top_idx[t]; top_idx[t] = top_idx[ixj]; top_idx[ixj] = i0;
                }
            }
            __syncthreads();
        }
    }

    // ---------------- async-stage loc + anchor rows of the top-200 into LDS ----------------
    // gfx1250 GLOBAL_LOAD_ASYNC_TO_LDS_B128: per-lane 64-bit global addr -> per-lane LDS addr,
    // tracked by ASYNCcnt (drained with s_wait_asynccnt).
    if (t < TOPK) {
        int a = top_idx[t];
        const float* gl = loc + ((size_t)b * A + (size_t)a) * 4;   // 16B-aligned row
        const float* ga = anchors + (size_t)a * 4;                 // 16B-aligned row
        unsigned dl = (unsigned)(size_t)(&lds_loc[t * 4]);         // LDS byte offset = addr[31:0]
        unsigned da = (unsigned)(size_t)(&lds_anc[t * 4]);
        asm volatile("global_load_async_to_lds_b128 %0, %1, off"
                     :: "v"(dl), "v"(gl) : "memory");
        asm volatile("global_load_async_to_lds_b128 %0, %1, off"
                     :: "v"(da), "v"(ga) : "memory");
    }
    asm volatile("s_wait_asynccnt 0x0" ::: "memory");
    __syncthreads();

    // ---------------- decode surviving boxes (SSD center-variance) ----------------
    if (t < PADK) {
        if (t < TOPK) {
            float l0 = lds_loc[t * 4 + 0], l1 = lds_loc[t * 4 + 1];
            float l2 = lds_loc[t * 4 + 2], l3 = lds_loc[t * 4 + 3];
            float pcx = lds_anc[t * 4 + 0], pcy = lds_anc[t * 4 + 1];
            float pw  = lds_anc[t * 4 + 2], ph  = lds_anc[t * 4 + 3];
            float cx = pcx + l0 * VAR0 * pw;
            float cy = pcy + l1 * VAR0 * ph;
            float w  = pw * __expf(l2 * VAR1);
            float h  = ph * __expf(l3 * VAR1);
            bbx[t] = make_float4(cx - 0.5f * w, cy - 0.5f * h,
                                 cx + 0.5f * w, cy + 0.5f * h);
            barea[t]  = w * h;
            bscore[t] = __uint_as_float(top_key[t]);
        } else {
            bbx[t] = make_float4(0.f, 0.f, 0.f, 0.f);
            barea[t] = 0.f; bscore[t] = 0.f;
        }
    }
    // zero suppression masks + keep words
    for (int i = t; i < PADK * 7; i += NT) (&sup[0][0])[i] = 0u;
    if (t < 7) keepw[t] = 0u;
    __syncthreads();
    if (t < TOPK && bscore[t] > CONF_T)
        atomicOr(&keepw[t >> 5], 1u << (t & 31));
    __syncthreads();

    // ---------------- pairwise IoU via WMMA area outer-sum + VALU inter ----------------
    // union_base[i][j] = area_i + area_j on the matrix pipe:
    //   D = A(16x4) * B(4x16), A cols = (area_i, 1, 0, 0), B rows = (1, area_j, 0, 0)
    {
        const int lane = t & 31;
        const int wave = t >> 5;
        const int ml   = lane & 15;
        const bool lo  = lane < 16;       // lanes 0-15 carry K=0,1 per 16x4 f32 A layout
        #pragma unroll 2
        for (int tile = wave; tile < NTILE * NTILE; tile += 8) {
            const int ti = (tile / NTILE) * 16;
            const int tj = (tile % NTILE) * 16;
            v2f av, bv;
            av[0] = lo ? barea[ti + ml] : 0.f;   // K=0 column: area_i
            av[1] = lo ? 1.f            : 0.f;   // K=1 column: ones
            bv[0] = lo ? 1.f            : 0.f;   // K=0 row: ones
            bv[1] = lo ? barea[tj + ml] : 0.f;   // K=1 row: area_j
            v8f cz = {0.f, 0.f, 0.f, 0.f, 0.f, 0.f, 0.f, 0.f};
            v8f d = __builtin_amdgcn_wmma_f32_16x16x4_f32(
                false, av, false, bv, (short)0, cz, false, false);
            // fixed column n for this lane across all 8 D rows
            const int n = tj + (lane & 15);
            const float4 bn = bbx[n];            // single ds_load_b128
            const int mbase = ti + ((lane >> 4) << 3);
            #pragma unroll
            for (int v = 0; v < 8; ++v) {
                const int m = mbase + v;         // C/D layout: VGPR v -> M=v / v+8
                const float4 bm = bbx[m];        // single ds_load_b128
                float iw = fminf(bm.z, bn.z) - fmaxf(bm.x, bn.x);
                float ih = fminf(bm.w, bn.w) - fmaxf(bm.y, bn.y);
                iw = fmaxf(iw, 0.f); ih = fmaxf(ih, 0.f);
                float inter = iw * ih;
                float uni   = d[v] - inter + 1e-9f;   // area_i + area_j - inter
                // predicate only the LDS atomic, keep the math branch-free
                if ((n > m) & (inter > NMS_T * uni))
                    atomicOr(&sup[m][n >> 5], 1u << (n & 31));
            }
        }
    }
    __syncthreads();

    // ---------------- sequential greedy NMS on bitmasks ----------------
    for (int i = 0; i < TOPK; ++i) {
        bool ki = (keepw[i >> 5] >> (i & 31)) & 1u;
        __syncthreads();
        if (ki && t < 7) keepw[t] &= ~sup[i][t];
        __syncthreads();
    }

    // ---------------- emit [B,C,200,4] boxes then [B,C,200,1] scores ----------------
    if (t < TOPK) {
        bool kp = (keepw[t >> 5] >> (t & 31)) & 1u;
        float4 bo = bbx[t];
        size_t ob = (size_t)bc * TOPK + t;
        out[ob * 4 + 0] = kp ? bo.x : 0.f;
        out[ob * 4 + 1] = kp ? bo.y : 0.f;
        out[ob * 4 + 2] = kp ? bo.z : 0.f;
        out[ob * 4 + 3] = kp ? bo.w : 0.f;
        out[(size_t)B * C * TOPK * 4 + ob] = kp ? bscore[t] : 0.f;
    }
}

extern "C" void kernel_launch(void* const* d_in, const int* in_sizes, int n_in,
                              void* d_out, int out_size, void* d_ws, size_t ws_size,
                              hipStream_t stream) {
    const float* loc     = (const float*)d_in[0];   // [B, A, 4]
    const float* conf    = (const float*)d_in[1];   // [B, A*C]
    const float* anchors = (const float*)d_in[2];   // [A, 4]
    float* out = (float*)d_out;

    const int A = in_sizes[2] / 4;
    const int B = in_sizes[0] / (A * 4);
    const int C = in_sizes[1] / (B * A);

    dim3 grid(B * C);   // 648 independent (image, class) NMS problems
    dim3 block(NT);
    detect_nms_kernel<<<grid, block, 0, stream>>>(loc, conf, anchors, out, B, A, C);
}